// MoE_48223892799904
// MI455X (gfx1250) — compile-verified
//
#include <hip/hip_runtime.h>

// ---------------------------------------------------------------------------
// MoE (16 experts, top-2) + shared expert + load-balance loss for gfx1250.
// fp32 WMMA (v_wmma_f32_16x16x4_f32): kernel is HBM-bound (~350MB @ 23.3TB/s),
// so full-precision matrix ops cost nothing vs bf16 and match the reference.
// Round 3: fix async-load builtin arg types (param is v4i addrspace ptr, per
// hipcc diagnostic). Double-buffered async global->LDS staging of Wfc slabs
// (GLOBAL_LOAD_ASYNC_TO_LDS_B128 + s_wait_asynccnt) with sync-b128 fallback.
// ---------------------------------------------------------------------------

#define D_    1024
#define H_    2048
#define E_    16
#define T_    8192
#define KTOP  2
#define ALPHA 0.001f

typedef __attribute__((ext_vector_type(2))) float v2f;
typedef __attribute__((ext_vector_type(8))) float v8f;
typedef int v4i_vs __attribute__((vector_size(16)));   // matches builtin param

#if __has_builtin(__builtin_amdgcn_global_load_async_to_lds_b128) && \
    __has_builtin(__builtin_amdgcn_s_wait_asynccnt)
#define HAVE_ASYNC_LDS 1
#else
#define HAVE_ASYNC_LDS 0
#endif

__device__ __forceinline__ void stage_b128(float* lds_dst, const float* gsrc) {
#if HAVE_ASYNC_LDS
    typedef v4i_vs __attribute__((address_space(1))) *gp_t;   // global v4i*
    typedef v4i_vs __attribute__((address_space(3))) *lp_t;   // LDS    v4i*
    __builtin_amdgcn_global_load_async_to_lds_b128(
        (gp_t)gsrc, (lp_t)lds_dst, 0, 0);
#else
    *(float4*)lds_dst = *(const float4*)gsrc;
#endif
}

__device__ __forceinline__ float silu_f(float v) {
    // v * rcp(1+exp(-v)): v_rcp_f32 (~1 ulp) instead of IEEE divide expansion
    return v * __builtin_amdgcn_rcpf(1.0f + __expf(-v));
}

// ---------------------------------------------------------------------------
// Zero the atomic counters in workspace (harness does not re-zero between runs)
// ---------------------------------------------------------------------------
__global__ void init_ws_kernel(int* counts, float* prob_sum) {
    int i = threadIdx.x;
    if (i < E_) { counts[i] = 0; prob_sum[i] = 0.0f; }
}

// ---------------------------------------------------------------------------
// Gate: wave-per-token. scores = softmax(x @ gate_w); top-2; scatter token ids
// and combine weights into per-expert lists; accumulate loss statistics.
// ---------------------------------------------------------------------------
__global__ void gate_kernel(const float* __restrict__ x,
                            const float* __restrict__ gw,
                            const float* __restrict__ gb,
                            int* __restrict__ counts,
                            float* __restrict__ prob_sum,
                            int* __restrict__ tok_list,
                            float* __restrict__ wt_list) {
    __shared__ float gws[D_ * E_];
    __shared__ float probs[E_];
    const int tid = threadIdx.x;

    for (int i = tid; i < (D_ * E_) / 4; i += blockDim.x)
        ((float4*)gws)[i] = ((const float4*)gw)[i];
    if (tid < E_) probs[tid] = 0.0f;
    __syncthreads();

    const int wave = tid >> 5;
    const int lane = tid & 31;
    const int t    = blockIdx.x * 8 + wave;
    const float* xt = x + (size_t)t * D_;

    float acc[E_];
#pragma unroll
    for (int e = 0; e < E_; ++e) acc[e] = 0.0f;

    for (int d = lane; d < D_; d += 32) {
        const float xv = xt[d];
        const float* g = &gws[d * E_];
#pragma unroll
        for (int e = 0; e < E_; ++e) acc[e] = fmaf(xv, g[e], acc[e]);
    }

    float sval = 0.0f;
#pragma unroll
    for (int e = 0; e < E_; ++e) {
        float v = acc[e];
#pragma unroll
        for (int off = 16; off > 0; off >>= 1) v += __shfl_xor(v, off, 32);
        if (lane == e) sval = v;
    }

    // softmax across lanes 0..15 (xor offsets < 16 keep 16-groups separate)
    float m = sval;
#pragma unroll
    for (int off = 8; off > 0; off >>= 1) m = fmaxf(m, __shfl_xor(m, off, 32));
    float p = __expf(sval - m);
    float s = p;
#pragma unroll
    for (int off = 8; off > 0; off >>= 1) s += __shfl_xor(s, off, 32);
    const float score  = p / s;
    const float biased = score + gb[lane & 15];

    float m1 = biased;
#pragma unroll
    for (int off = 8; off > 0; off >>= 1) m1 = fmaxf(m1, __shfl_xor(m1, off, 32));
    int c1 = (biased == m1) ? (lane & 15) : 999;
#pragma unroll
    for (int off = 8; off > 0; off >>= 1) { int o = __shfl_xor(c1, off, 32); c1 = (o < c1) ? o : c1; }

    float v2 = ((lane & 15) == c1) ? -3.0e38f : biased;
    float m2 = v2;
#pragma unroll
    for (int off = 8; off > 0; off >>= 1) m2 = fmaxf(m2, __shfl_xor(m2, off, 32));
    int c2 = (v2 == m2) ? (lane & 15) : 999;
#pragma unroll
    for (int off = 8; off > 0; off >>= 1) { int o = __shfl_xor(c2, off, 32); c2 = (o < c2) ? o : c2; }

    const float w1 = __shfl(score, c1, 32);
    const float w2 = __shfl(score, c2, 32);

    if (lane == 0) {
        int s1 = atomicAdd(&counts[c1], 1);
        tok_list[c1 * T_ + s1] = t;  wt_list[c1 * T_ + s1] = w1;
        int s2 = atomicAdd(&counts[c2], 1);
        tok_list[c2 * T_ + s2] = t;  wt_list[c2 * T_ + s2] = w2;
    }
    if (lane < 16) atomicAdd(&probs[lane], score);
    __syncthreads();
    if (tid < E_) atomicAdd(&prob_sum[tid], probs[tid]);
}

// ---------------------------------------------------------------------------
// Fused MLP: y(+)= silu(x_tile @ Wfc + bfc) @ Wproj + bproj, scaled by combine
// weight. 16-token tile per block, 8 waves (256 threads, wave32).
//   - x tile in LDS (padded stride 1028 -> conflict-free b64 A reads)
//   - H in chunks of 128: GEMM1 -> silu -> h in LDS -> GEMM2
//   - GEMM1 B slabs (64 K-rows x 128 cols) async-DMA'd into a double-buffered
//     LDS slab (stride 144: low half-wave banks {c..c+15}, high half {c+32..})
//   - GEMM2 B streamed from global (L2-resident expert weights)
//   - All matrix math via v_wmma_f32_16x16x4_f32 (v8f accumulators)
// ---------------------------------------------------------------------------
template <bool SHARED>
__global__ void mlp_kernel(const float* __restrict__ x,
                           const float* __restrict__ wfc,
                           const float* __restrict__ bfc,
                           const float* __restrict__ wproj,
                           const float* __restrict__ bproj,
                           const int*  __restrict__ counts,
                           const int*  __restrict__ tok_list,
                           const float* __restrict__ wt_list,
                           float* __restrict__ out) {
    const int tile = blockIdx.x;
    const int e    = SHARED ? 0 : blockIdx.y;
    int cnt = 0;
    if (!SHARED) {
        cnt = counts[e];
        if (tile * 16 >= cnt) return;    // block-uniform: EXEC stays all-ones
    }

    __shared__ float xs[16 * 1028];      // 16 x 1024 fp32, padded rows
    __shared__ float hs[16 * 132];       // 16 x 128 fp32 h-chunk, padded rows
    __shared__ float wbuf[2][64 * 144];  // double-buffered Wfc slab (64K x 128N)
    __shared__ float wts[16];
    __shared__ int   toks[16];

    const int tid = threadIdx.x;
    if (tid < 16) {
        if (SHARED) { toks[tid] = tile * 16 + tid; wts[tid] = 1.0f; }
        else {
            int g = tile * 16 + tid;
            if (g < cnt) { toks[tid] = tok_list[e * T_ + g]; wts[tid] = wt_list[e * T_ + g]; }
            else         { toks[tid] = 0;                    wts[tid] = 0.0f; }
        }
    }
    __syncthreads();

    for (int v = tid; v < 16 * (D_ / 4); v += blockDim.x) {
        const int r = v >> 8;
        const int c = v & 255;
        float4 val = *(const float4*)&x[(size_t)toks[r] * D_ + c * 4];
        *(float4*)&xs[r * 1028 + c * 4] = val;
    }
    __syncthreads();

    const float* Wfc = SHARED ? wfc   : wfc   + (size_t)e * D_ * H_;
    const float* Bfc = SHARED ? bfc   : bfc   + (size_t)e * H_;
    const float* Wpj = SHARED ? wproj : wproj + (size_t)e * H_ * D_;
    const float* Bpj = SHARED ? bproj : bproj + (size_t)e * D_;

    const int wave = tid >> 5;
    const int lane = tid & 31;
    const int nn   = lane & 15;
    const int hh   = lane >> 4;

    v8f yacc[8];
#pragma unroll
    for (int i = 0; i < 8; ++i)
        yacc[i] = (v8f){0.f, 0.f, 0.f, 0.f, 0.f, 0.f, 0.f, 0.f};
    const int yc0 = wave * 128;

    for (int ch = 0; ch < H_ / 128; ++ch) {
        const int hc = ch * 128;

        // Stage a 64x128 slab of Wfc: 2048 b128 chunks, 8 per thread per slab.
        auto stage_slab = [&](int s, int buf) {
            const float* src = Wfc + (size_t)(s * 64) * H_ + hc;
#pragma unroll
            for (int i = 0; i < 8; ++i) {
                const int v   = tid + i * 256;
                const int row = v >> 5;
                const int c4  = (v & 31) << 2;
                stage_b128(&wbuf[buf][row * 144 + c4],
                           src + (size_t)row * H_ + c4);
            }
        };

        // ---- GEMM1: h[:, hc+wave*16 .. +16) = x_tile @ Wfc, K = 1024 ----
        const int nc1 = wave * 16;
        v8f hacc = (v8f){0.f, 0.f, 0.f, 0.f, 0.f, 0.f, 0.f, 0.f};

        stage_slab(0, 0);                          // prologue
        for (int s = 0; s < 16; ++s) {             // 16 slabs x 64 K-rows
            const int buf = s & 1;
            if (s + 1 < 16) {
                stage_slab(s + 1, (s + 1) & 1);    // prefetch next slab
#if HAVE_ASYNC_LDS
                __builtin_amdgcn_s_wait_asynccnt(8);   // this slab's 8 DMAs done
#endif
            } else {
#if HAVE_ASYNC_LDS
                __builtin_amdgcn_s_wait_asynccnt(0);
#endif
            }
            __syncthreads();                       // slab visible to all waves
            const float* wb = &wbuf[buf][0];
#pragma unroll
            for (int kk = 0; kk < 64; kk += 4) {
                v2f a = *(const v2f*)&xs[nn * 1028 + s * 64 + kk + 2 * hh];
                v2f b;
                b.x = wb[(kk + 2 * hh)     * 144 + nc1 + nn];
                b.y = wb[(kk + 2 * hh + 1) * 144 + nc1 + nn];
                hacc = __builtin_amdgcn_wmma_f32_16x16x4_f32(
                    false, a, false, b, (short)0, hacc, false, false);
            }
            __syncthreads();                       // buf reusable next+1 slab
        }

        const float bf = Bfc[hc + wave * 16 + nn];
#pragma unroll
        for (int r = 0; r < 8; ++r) {
            hs[(r + 8 * hh) * 132 + wave * 16 + nn] = silu_f(hacc[r] + bf);
        }
        __syncthreads();

        // ---- GEMM2: y[:, yc0 .. +128) += h_chunk @ Wproj, K = 128 ----
#pragma unroll
        for (int t8 = 0; t8 < 8; ++t8) {
            const int col = yc0 + t8 * 16;
            const float* wp = Wpj + (size_t)hc * D_ + col + nn;
            v8f c = yacc[t8];
            for (int kb = 0; kb < 128; kb += 4) {
                v2f a = *(const v2f*)&hs[nn * 132 + kb + 2 * hh];
                v2f b;
                b.x = wp[(size_t)(kb + 2 * hh)     * D_];
                b.y = wp[(size_t)(kb + 2 * hh + 1) * D_];
                c = __builtin_amdgcn_wmma_f32_16x16x4_f32(
                    false, a, false, b, (short)0, c, false, false);
            }
            yacc[t8] = c;
        }
        __syncthreads();                           // protect hs for next chunk
    }

    // ---- Epilogue: add bproj, scale by combine weight, write out ----
#pragma unroll
    for (int t8 = 0; t8 < 8; ++t8) {
        const int col = yc0 + t8 * 16 + nn;
        const float bp = Bpj[col];
#pragma unroll
        for (int r = 0; r < 8; ++r) {
            const int mrow = r + 8 * hh;
            const float val = (yacc[t8][r] + bp) * wts[mrow];
            const size_t o = (size_t)toks[mrow] * D_ + col;
            if (SHARED) out[o] = val;
            else        atomicAdd(&out[o], val);
        }
    }
}

// ---------------------------------------------------------------------------
// Load-balance loss: ALPHA * sum_e (count_e * E / (K*T)) * mean_prob_e
// ---------------------------------------------------------------------------
__global__ void loss_kernel(const int* __restrict__ counts,
                            const float* __restrict__ prob_sum,
                            float* __restrict__ out_loss) {
    const int lane = threadIdx.x;
    float part = 0.0f;
    if (lane < E_) {
        float f  = (float)counts[lane] * (float)E_ /
                   ((float)(KTOP * T_) + 1e-6f);
        float pm = prob_sum[lane] / (float)T_;
        part = f * pm;
    }
#pragma unroll
    for (int off = 16; off > 0; off >>= 1) part += __shfl_xor(part, off, 32);
    if (lane == 0) out_loss[0] = ALPHA * part;
}

// ---------------------------------------------------------------------------
extern "C" void kernel_launch(void* const* d_in, const int* in_sizes, int n_in,
                              void* d_out, int out_size, void* d_ws, size_t ws_size,
                              hipStream_t stream) {
    const float* x      = (const float*)d_in[0];
    const float* gw     = (const float*)d_in[1];
    const float* gb     = (const float*)d_in[2];
    const float* wfc    = (const float*)d_in[3];
    const float* bfc    = (const float*)d_in[4];
    const float* wproj  = (const float*)d_in[5];
    const float* bproj  = (const float*)d_in[6];
    const float* swfc   = (const float*)d_in[7];
    const float* sbfc   = (const float*)d_in[8];
    const float* swproj = (const float*)d_in[9];
    const float* sbproj = (const float*)d_in[10];
    float* out = (float*)d_out;

    char* ws        = (char*)d_ws;
    int*   counts   = (int*)ws;                                // 16 ints
    float* prob_sum = (float*)(ws + 64);                       // 16 floats
    int*   tok_list = (int*)(ws + 128);                        // E*T ints
    float* wt_list  = (float*)(ws + 128 + (size_t)E_ * T_ * 4);// E*T floats

    init_ws_kernel<<<1, 32, 0, stream>>>(counts, prob_sum);

    gate_kernel<<<T_ / 8, 256, 0, stream>>>(x, gw, gb, counts, prob_sum,
                                            tok_list, wt_list);

    // Shared expert: initializes out with plain stores
    mlp_kernel<true><<<T_ / 16, 256, 0, stream>>>(
        x, swfc, sbfc, swproj, sbproj, nullptr, nullptr, nullptr, out);

    // Routed experts: grid.x = tile (fast dim) so one expert's tiles dispatch
    // together -> 16MB of weights resident in L2 while its tiles run
    mlp_kernel<false><<<dim3(T_ / 16, E_), 256, 0, stream>>>(
        x, wfc, bfc, wproj, bproj, counts, tok_list, wt_list, out);

    loss_kernel<<<1, 32, 0, stream>>>(counts, prob_sum, out + (size_t)T_ * D_);
}